// MSALayer_55877524521421
// MI455X (gfx1250) — compile-verified
//
#include <hip/hip_runtime.h>

typedef float v2f __attribute__((ext_vector_type(2)));
typedef float v8f __attribute__((ext_vector_type(8)));

static constexpr int Mm = 32;     // MSA depth
static constexpr int Nn = 2048;   // nodes
static constexpr int Ee = 32768;  // edges
static constexpr int Hh = 128;    // hidden
static constexpr int Cc = 32;     // channels

// LDS layout constants (floats)
static constexpr int TS = 34;               // transposed tile stride [c][m]
static constexpr int TILE_F = 32 * TS;      // 1088 floats per tile
static constexpr int NBS = 1026;            // nb row stride (even -> b64 loads)
static constexpr int WLS = 132;             // wlds row stride (16B aligned)

__device__ __forceinline__ v8f wmma4(v2f a, v2f b, v8f c) {
  // D = A(16x4 f32) * B(4x16 f32) + C(16x16 f32)
  return __builtin_amdgcn_wmma_f32_16x16x4_f32(false, a, false, b, (short)0, c,
                                               false, false);
}

__device__ __forceinline__ unsigned lds_off(const void* p) {
  // generic pointer to LDS: low 32 bits are the wave-relative LDS address
  return (unsigned)(uintptr_t)p;
}

__device__ __forceinline__ void async_g2l_b32(unsigned ldst, const void* gsrc) {
  asm volatile("global_load_async_to_lds_b32 %0, %1, off"
               :
               : "v"(ldst), "v"(gsrc)
               : "memory");
}

__device__ __forceinline__ void async_g2l_b128(unsigned ldst, const void* gsrc) {
  asm volatile("global_load_async_to_lds_b128 %0, %1, off"
               :
               : "v"(ldst), "v"(gsrc)
               : "memory");
}

__device__ __forceinline__ void wait_async0() {
  asm volatile("s_wait_asynccnt 0x0" ::: "memory");
}

__device__ __forceinline__ void wait_ds0() {
  asm volatile("s_wait_dscnt 0x0" ::: "memory");
}

// ---------------------------------------------------------------------------
// Transpose w_lin (1024 x 128) -> wT (128 x 1024) once; enables contiguous
// async b128 staging in phase B.
// ---------------------------------------------------------------------------
__global__ void __launch_bounds__(256) transW(const float* __restrict__ wLin,
                                              float* __restrict__ wT) {
  const int idx = blockIdx.x * 256 + threadIdx.x;  // 131072 elements
  const int k = idx >> 7;
  const int h = idx & 127;
  wT[(size_t)h * 1024 + k] = wLin[idx];
}

// ---------------------------------------------------------------------------
// Phase A: a = hV @ w_a ; b = hV @ w_b   (M*N x H) @ (H x C)
// Weights staged transposed [c][k] so B operands are single ds_load_b64.
// ---------------------------------------------------------------------------
__global__ void __launch_bounds__(256) phaseA(
    const float* __restrict__ hV, const float* __restrict__ wA,
    const float* __restrict__ wB, float* __restrict__ aW,
    float* __restrict__ bW) {
  __shared__ float la[32 * 130];  // [c][k], stride 130 (even, bank advance 2)
  __shared__ float lb[32 * 130];
  const int tid = threadIdx.x;
  for (int i = tid; i < 128 * 32; i += 256) {
    const int k = i >> 5, c = i & 31;
    la[c * 130 + k] = wA[i];
    lb[c * 130 + k] = wB[i];
  }
  __syncthreads();

  const int lane = tid & 31;
  const int wv = tid >> 5;
  const int r0 = blockIdx.x * 128 + wv * 16;
  const int lr = lane & 15;
  const int kh = (lane >> 4) << 1;
  const float* rp = hV + (size_t)(r0 + lr) * Hh;

  v8f z = {0.f, 0.f, 0.f, 0.f, 0.f, 0.f, 0.f, 0.f};
  v8f accA0 = z, accA1 = z, accB0 = z, accB1 = z;

  for (int kk = 0; kk < 32; ++kk) {
    const int k0 = (kk << 2) + kh;
    const v2f av = *(const v2f*)(rp + k0);  // contiguous global b64
    const v2f ba0 = *(const v2f*)(la + lr * 130 + k0);
    const v2f ba1 = *(const v2f*)(la + (16 + lr) * 130 + k0);
    const v2f bb0 = *(const v2f*)(lb + lr * 130 + k0);
    const v2f bb1 = *(const v2f*)(lb + (16 + lr) * 130 + k0);
    accA0 = wmma4(av, ba0, accA0);
    accA1 = wmma4(av, ba1, accA1);
    accB0 = wmma4(av, bb0, accB0);
    accB1 = wmma4(av, bb1, accB1);
  }

  const int rbase = r0 + ((lane >> 4) << 3);
#pragma unroll
  for (int v = 0; v < 8; ++v) {
    const size_t row = (size_t)(rbase + v);
    aW[row * Cc + lr] = accA0[v];
    aW[row * Cc + 16 + lr] = accA1[v];
    bW[row * Cc + lr] = accB0[v];
    bW[row * Cc + 16 + lr] = accB1[v];
  }
}

// ---------------------------------------------------------------------------
// Phase B: fused per-edge outer-product + projection + residual + LayerNorm.
// 32 edges per 256-thread workgroup (8 waves, 4 edges/wave in stage 1,
// 2 row-tiles x 1 col-tile per wave in stage 2 -> B operand reused 2x).
// ---------------------------------------------------------------------------
__global__ void __launch_bounds__(256) phaseB(
    const float* __restrict__ aW, const float* __restrict__ bW,
    const long long* __restrict__ eidx, const float* __restrict__ mask,
    const float* __restrict__ wT, const float* __restrict__ bLin,
    const float* __restrict__ hE0, const float* __restrict__ lnW,
    const float* __restrict__ lnB, float* __restrict__ outE) {
  extern __shared__ float sm[];
  // region 0: per-wave transposed edge tiles (stage 1); overlaid by wlds in
  // stage 2 (wlds = 128*132 = 16896 floats <= 2*8*1088 = 17408 floats)
  float* at = sm;                       // 8 waves * [32 c][34 m]
  float* bt = at + 8 * TILE_F;          // 8 waves * [32 c][34 m]
  float* wlds = sm;                     // stage-2 overlay: [128 h][132 k]
  float* nb = sm + 2 * 8 * TILE_F;      // 32 edges * NBS
  float* xb = nb + 32 * NBS;            // 32 * 128 pre-LN

  const int tid = threadIdx.x;
  const int lane = tid & 31;
  const int wv = tid >> 5;
  const int eBase = blockIdx.x * 32;
  const int lr = lane & 15;
  const int kh = (lane >> 4) << 1;
  float* atw = at + wv * TILE_F;
  float* btw = bt + wv * TILE_F;

  v8f z = {0.f, 0.f, 0.f, 0.f, 0.f, 0.f, 0.f, 0.f};

  // ---- Stage 1: each wave does 4 edges: P = B^T (diag(mask) A) ------------
  for (int sub = 0; sub < 4; ++sub) {
    const int slot = wv * 4 + sub;
    const int e = eBase + slot;
    const int src = (int)eidx[e];
    const int tgt = (int)eidx[Ee + e];

    // lane == m: per-row mask; wave-reduce for denom
    const float ms = mask[lane * Nn + src] * mask[lane * Nn + tgt];
    float s = ms;
#pragma unroll
    for (int off = 16; off > 0; off >>= 1) s += __shfl_xor(s, off);
    const float invd = 1.0f / fmaxf(s, 1.0f);

    wait_ds0();  // previous sub's LDS reads must finish before overwrite

    // stage tiles transposed [c][m]; b-tile via async global->LDS scatter
    const unsigned btw_base = lds_off(btw + lane * TS);
#pragma unroll 4
    for (int j = 0; j < 32; ++j) {
      const float mj = __shfl(ms, j);
      atw[lane * TS + j] = aW[(size_t)(j * Nn + src) * Cc + lane] * mj;
      async_g2l_b32(btw_base + j * 4u,
                    bW + (size_t)(j * Nn + tgt) * Cc + lane);
    }
    wait_ds0();     // a-tile LDS stores visible
    wait_async0();  // b-tile async writes visible

    v8f p00 = z, p01 = z, p10 = z, p11 = z;
#pragma unroll
    for (int kk = 0; kk < 8; ++kk) {
      const int m0 = (kk << 2) + kh;  // k dim = MSA rows m
      const v2f aB0 = *(const v2f*)(btw + lr * TS + m0);
      const v2f aB1 = *(const v2f*)(btw + (16 + lr) * TS + m0);
      const v2f bA0 = *(const v2f*)(atw + lr * TS + m0);
      const v2f bA1 = *(const v2f*)(atw + (16 + lr) * TS + m0);
      p00 = wmma4(aB0, bA0, p00);
      p01 = wmma4(aB0, bA1, p01);
      p10 = wmma4(aB1, bA0, p10);
      p11 = wmma4(aB1, bA1, p11);
    }

    // P[beta][alpha] -> nb[slot][beta*32+alpha], scaled by 1/denom
    float* nrow = nb + slot * NBS;
    const int bOff = (lane >> 4) << 3;
#pragma unroll
    for (int v = 0; v < 8; ++v) {
      const int br = v + bOff;
      nrow[br * 32 + lr] = p00[v] * invd;
      nrow[br * 32 + 16 + lr] = p01[v] * invd;
      nrow[(br + 16) * 32 + lr] = p10[v] * invd;
      nrow[(br + 16) * 32 + 16 + lr] = p11[v] * invd;
    }
  }

  // ---- Stage 2: (32 x 1024) @ w_lin(1024 x 128), K chunked through LDS ----
  const int hcol = wv * 16 + lr;
  v8f acc0 = z, acc1 = z;
  const unsigned wlds_base = lds_off(wlds);

  for (int ch = 0; ch < 8; ++ch) {
    __syncthreads();  // all waves done reading previous chunk / stage-1 tiles
    // async b128 stage: wT[h][ch*128 ..+128) -> wlds[h][0..128)
    for (int i = tid; i < 4096; i += 256) {  // 4096 16B blocks
      const int h = i >> 5;
      const int kb = i & 31;
      async_g2l_b128(wlds_base + (unsigned)(h * WLS + kb * 4) * 4u,
                     wT + (size_t)h * 1024 + ch * 128 + kb * 4);
    }
    wait_async0();
    __syncthreads();

    const float* nb0 = nb + lr * NBS + ch * 128;
    const float* nb1 = nb + (16 + lr) * NBS + ch * 128;
    const float* wl = wlds + hcol * WLS;
    for (int kk = 0; kk < 32; ++kk) {
      const int k0 = (kk << 2) + kh;
      const v2f b2 = *(const v2f*)(wl + k0);
      const v2f a0 = *(const v2f*)(nb0 + k0);
      const v2f a1 = *(const v2f*)(nb1 + k0);
      acc0 = wmma4(a0, b2, acc0);
      acc1 = wmma4(a1, b2, acc1);
    }
  }

  const float bl = bLin[hcol];
  const int sOff = (lane >> 4) << 3;
#pragma unroll
  for (int v = 0; v < 8; ++v) {
    const int s0 = v + sOff;
    const int s1 = 16 + v + sOff;
    xb[s0 * 128 + hcol] = acc0[v] + bl + hE0[(size_t)(eBase + s0) * Hh + hcol];
    xb[s1 * 128 + hcol] = acc1[v] + bl + hE0[(size_t)(eBase + s1) * Hh + hcol];
  }
  __syncthreads();

  // ---- LayerNorm over H=128: 8 waves x 4 rows ----------------------------
#pragma unroll
  for (int rr = 0; rr < 4; ++rr) {
    const int r = wv * 4 + rr;
    const float v0 = xb[r * 128 + lane];
    const float v1 = xb[r * 128 + lane + 32];
    const float v2 = xb[r * 128 + lane + 64];
    const float v3 = xb[r * 128 + lane + 96];
    float s1 = v0 + v1 + v2 + v3;
    float s2 = v0 * v0 + v1 * v1 + v2 * v2 + v3 * v3;
#pragma unroll
    for (int off = 16; off > 0; off >>= 1) {
      s1 += __shfl_xor(s1, off);
      s2 += __shfl_xor(s2, off);
    }
    const float mu = s1 * (1.0f / 128.0f);
    const float var = s2 * (1.0f / 128.0f) - mu * mu;
    const float rs = rsqrtf(var + 1e-5f);
    float* orow = outE + (size_t)(eBase + r) * Hh;
    orow[lane] = (v0 - mu) * rs * lnW[lane] + lnB[lane];
    orow[lane + 32] = (v1 - mu) * rs * lnW[lane + 32] + lnB[lane + 32];
    orow[lane + 64] = (v2 - mu) * rs * lnW[lane + 64] + lnB[lane + 64];
    orow[lane + 96] = (v3 - mu) * rs * lnW[lane + 96] + lnB[lane + 96];
  }
}

// ---------------------------------------------------------------------------
extern "C" void kernel_launch(void* const* d_in, const int* in_sizes, int n_in,
                              void* d_out, int out_size, void* d_ws,
                              size_t ws_size, hipStream_t stream) {
  const float* hV = (const float*)d_in[0];            // (M,N,H)
  const float* hE = (const float*)d_in[1];            // (M,E,H), slice 0 used
  const long long* eidx = (const long long*)d_in[2];  // (2,E) int64
  const float* mask = (const float*)d_in[3];          // (M,N)
  const float* wA = (const float*)d_in[4];            // (H,C)
  const float* wB = (const float*)d_in[5];            // (H,C)
  const float* wLin = (const float*)d_in[6];          // (C*C,H)
  const float* bLin = (const float*)d_in[7];          // (H,)
  const float* lnW = (const float*)d_in[8];
  const float* lnB = (const float*)d_in[9];

  float* out = (float*)d_out;             // h_V (N*H) then h_E (E*H)
  float* aW = (float*)d_ws;               // (M*N, C)
  float* bW = aW + (size_t)Mm * Nn * Cc;  // (M*N, C)
  float* wT = bW + (size_t)Mm * Nn * Cc;  // (H, C*C) transposed w_lin

  // h_V = h_V_msa[0]
  hipMemcpyAsync(out, hV, (size_t)Nn * Hh * sizeof(float),
                 hipMemcpyDeviceToDevice, stream);

  transW<<<512, 256, 0, stream>>>(wLin, wT);
  phaseA<<<(Mm * Nn) / 128, 256, 0, stream>>>(hV, wA, wB, aW, bW);

  const size_t smB =
      (size_t)(2 * 8 * TILE_F + 32 * NBS + 32 * 128) * sizeof(float);
  phaseB<<<Ee / 32, 256, smB, stream>>>(aW, bW, eidx, mask, wT, bLin, hE, lnW,
                                        lnB, out + (size_t)Nn * Hh);
}